// LGCNEncoder_75892072120406
// MI455X (gfx1250) — compile-verified
//
#include <hip/hip_runtime.h>
#include <hip/hip_bf16.h>

typedef __attribute__((ext_vector_type(2))) float v2f;
typedef __attribute__((ext_vector_type(8))) float v8f;

// ---------------- init: acc = x = concat(user_emb, item_emb) ----------------
__global__ void lgcn_init_kernel(const float* __restrict__ ue, const float* __restrict__ ie,
                                 float* __restrict__ x, float* __restrict__ acc,
                                 int u64, int ne) {
    int g = blockIdx.x * blockDim.x + threadIdx.x;
    if (g >= ne) return;
    float v = (g < u64) ? ue[g] : ie[g - u64];
    x[g] = v;
    acc[g] = v;
}

// ---------------- zero y ----------------
__global__ void lgcn_zero_kernel(float4* __restrict__ y, int ne4) {
    int g = blockIdx.x * blockDim.x + threadIdx.x;
    if (g >= ne4) return;
    y[g] = make_float4(0.f, 0.f, 0.f, 0.f);
}

// ---------------- SpMM: y[rows[e]] += vals[e] * x[cols[e]], one wave32/edge ----------------
__global__ void lgcn_spmm_kernel(const float* __restrict__ vals, const int* __restrict__ rows,
                                 const int* __restrict__ cols, const float* __restrict__ x,
                                 float* __restrict__ y, int nnz) {
    int e = blockIdx.x * 8 + (threadIdx.x >> 5);   // 8 waves / 256-thread block
    if (e >= nnz) return;
    int lane = threadIdx.x & 31;
    float v = vals[e];
    int r = rows[e];
    int c = cols[e];
    float2 xv = ((const float2*)(x + (size_t)c * 64))[lane];   // coalesced 256B row gather
    float* yp = y + (size_t)r * 64 + lane * 2;
    atomicAdd(yp,     v * xv.x);                               // global_atomic_add_f32, hits L2
    atomicAdd(yp + 1, v * xv.y);
}

// ---------------- acc += y ----------------
__global__ void lgcn_accadd_kernel(float4* __restrict__ acc, const float4* __restrict__ y, int ne4) {
    int g = blockIdx.x * blockDim.x + threadIdx.x;
    if (g >= ne4) return;
    float4 a = acc[g];
    float4 b = y[g];
    a.x += b.x; a.y += b.y; a.z += b.z; a.w += b.w;
    acc[g] = a;
}

// ---------------- u = acc[users]/4 ; also zero the loss scalar ----------------
__global__ void lgcn_gatheru_kernel(const float* __restrict__ acc, const int* __restrict__ users,
                                    float* __restrict__ outu, float* __restrict__ lossPtr, int b64) {
    int g = blockIdx.x * blockDim.x + threadIdx.x;
    if (g == 0) *lossPtr = 0.f;
    if (g >= b64) return;
    int b = g >> 6, k = g & 63;
    outu[g] = acc[(size_t)users[b] * 64 + k] * 0.25f;
}

// ---------------- VQ prep: zero-padded codebook copy + 0.5*||c||^2 table ----------------
// One 64-thread block per padded row.
__global__ void lgcn_vqprep_kernel(const float* __restrict__ cb, float* __restrict__ pcb,
                                   float* __restrict__ cn, int ncode) {
    int r = blockIdx.x;
    int k = threadIdx.x;
    float v = (r < ncode) ? cb[(size_t)r * 64 + k] : 0.f;
    pcb[(size_t)r * 64 + k] = v;
    __shared__ float s[64];
    s[k] = v * v;
    __syncthreads();
    if (k == 0) {
        float t = 0.f;
#pragma unroll
        for (int i = 0; i < 64; ++i) t += s[i];
        cn[r] = (r < ncode) ? 0.5f * t : __builtin_inff();
    }
}

// order-preserving f32 -> u32 (total order; larger float -> larger uint)
__device__ __forceinline__ unsigned f32_okey(float f) {
    unsigned u = __float_as_uint(f);
    return u ^ (unsigned)(((int)u >> 31) | 0x80000000);
}

// ---------------- VQ: scores via V_WMMA_F32_16X16X4_F32, argmin, loss, item_out ----------------
__global__ __launch_bounds__(256) void lgcn_vq_kernel(
    const float* __restrict__ acc, const float* __restrict__ pcb,
    const float* __restrict__ gCn, const float* __restrict__ cb,
    const int* __restrict__ items, float* __restrict__ outi,
    float* __restrict__ lossPtr, int U_, int ntiles, int B_, float lossScale) {
    int lane = threadIdx.x & 31;
    int wave = threadIdx.x >> 5;
    int tile = blockIdx.x * 8 + wave;
    int row0 = tile * 16;
    if (row0 >= B_) return;                         // wave-uniform exit
    int m15 = lane & 15;
    int hi  = lane >> 4;

    // Preload A tile: 16 rows of it = acc[U+items]/4.  Lane layout (ISA 16x4 f32 A):
    // lane(m15,hi) holds A[m15][4*ks + 2*hi + {0,1}] for k-step ks.
    const float* itp = acc + (size_t)(U_ + items[row0 + m15]) * 64;
    v2f aReg[16];
#pragma unroll
    for (int ks = 0; ks < 16; ++ks) {
        float2 t = ((const float2*)itp)[ks * 2 + hi];
        aReg[ks].x = t.x * 0.25f;
        aReg[ks].y = t.y * 0.25f;
    }

    // packed (orderable score << 32) | (0xFFFF - col): max => argmax score,
    // smallest col on ties (matches jnp.argmin first-index semantics)
    unsigned long long bestPk[8];
#pragma unroll
    for (int r = 0; r < 8; ++r) bestPk[r] = 0ull;

    for (int ct = 0; ct < ntiles; ++ct) {
        int colg = ct * 16 + m15;
        const float2* cbp = (const float2*)(pcb + (size_t)colg * 64);  // zero-padded, no mask
        v8f cacc = {};
#pragma unroll
        for (int ks = 0; ks < 16; ++ks) {
            float2 t = cbp[ks * 2 + hi];
            v2f b2; b2.x = t.x; b2.y = t.y;
            // D = A*B + C, fp32 16x16x4
            cacc = __builtin_amdgcn_wmma_f32_16x16x4_f32(
                false, aReg[ks], false, b2, (short)0, cacc, false, false);
        }
        float cnv = gCn[colg];                      // +inf for pad cols -> score = -inf
        unsigned low = (unsigned)(0xFFFF - colg);
#pragma unroll
        for (int r = 0; r < 8; ++r) {               // row = r + 8*hi, col = m15
            unsigned long long pk =
                ((unsigned long long)f32_okey(cacc[r] - cnv) << 32) | low;
#pragma unroll
            for (int off = 8; off >= 1; off >>= 1) { // branchless max within 16-lane half
                unsigned long long opk = __shfl_xor(pk, off);
                pk = (opk > pk) ? opk : pk;
            }
            bestPk[r] = (pk > bestPk[r]) ? pk : bestPk[r];
        }
    }

    // item_out = it + 0.25*q ; loss += (it - q)^2
    float lsum = 0.f;
#pragma unroll
    for (int r = 0; r < 16; ++r) {
        unsigned long long pk = __shfl(bestPk[r & 7], (r >> 3) << 4);
        int idx = 0xFFFF - (int)(pk & 0xFFFFull);
        int b = row0 + r;
        const float2* ip = (const float2*)(acc + (size_t)(U_ + items[b]) * 64);
        const float2* qp = (const float2*)(cb + (size_t)idx * 64);
        float2 iv = ip[lane]; iv.x *= 0.25f; iv.y *= 0.25f;
        float2 qv = qp[lane];
        float2 o; o.x = iv.x + 0.25f * qv.x; o.y = iv.y + 0.25f * qv.y;
        ((float2*)outi)[(size_t)b * 32 + lane] = o;
        float dx = iv.x - qv.x, dy = iv.y - qv.y;
        lsum += dx * dx + dy * dy;
    }
#pragma unroll
    for (int off = 16; off >= 1; off >>= 1) lsum += __shfl_xor(lsum, off);
    if (lane == 0) atomicAdd(lossPtr, lsum * lossScale);
}

extern "C" void kernel_launch(void* const* d_in, const int* in_sizes, int n_in,
                              void* d_out, int out_size, void* d_ws, size_t ws_size,
                              hipStream_t stream) {
    const float* ue    = (const float*)d_in[0];
    const float* ie    = (const float*)d_in[1];
    const float* cb    = (const float*)d_in[2];
    const float* vals  = (const float*)d_in[3];
    const int*   rows  = (const int*)d_in[4];
    const int*   cols  = (const int*)d_in[5];
    const int*   users = (const int*)d_in[6];
    const int*   items = (const int*)d_in[7];

    int U_    = in_sizes[0] / 64;
    int I_    = in_sizes[1] / 64;
    int ncode = in_sizes[2] / 64;
    int nnz   = in_sizes[3];
    int B_    = in_sizes[6];
    int N_    = U_ + I_;
    size_t NE = (size_t)N_ * 64;          // 6.4M floats = 25.6MB per buffer

    int ntiles = (ncode + 15) >> 4;
    int npad   = ntiles * 16;             // 512 for ncode=500

    float* x   = (float*)d_ws;            // 3 buffers, 76.8MB total — fits L2 (192MB)
    float* y   = x + NE;
    float* acc = y + NE;
    float* pcb = acc + NE;                // zero-padded codebook [npad x 64]
    float* cn  = pcb + (size_t)npad * 64; // 0.5*||c||^2 [npad]

    float* outu    = (float*)d_out;
    float* outi    = outu + (size_t)B_ * 64;
    float* lossPtr = outi + (size_t)B_ * 64;

    int ne  = (int)NE;
    int ne4 = (int)(NE / 4);

    lgcn_init_kernel<<<(ne + 255) / 256, 256, 0, stream>>>(ue, ie, x, acc, U_ * 64, ne);
    lgcn_vqprep_kernel<<<npad, 64, 0, stream>>>(cb, pcb, cn, ncode);
    for (int l = 0; l < 3; ++l) {
        lgcn_zero_kernel<<<(ne4 + 255) / 256, 256, 0, stream>>>((float4*)y, ne4);
        lgcn_spmm_kernel<<<(nnz + 7) / 8, 256, 0, stream>>>(vals, rows, cols, x, y, nnz);
        lgcn_accadd_kernel<<<(ne4 + 255) / 256, 256, 0, stream>>>((float4*)acc, (const float4*)y, ne4);
        float* t = x; x = y; y = t;
    }
    int b64 = B_ * 64;
    lgcn_gatheru_kernel<<<(b64 + 255) / 256, 256, 0, stream>>>(acc, users, outu, lossPtr, b64);
    float lossScale = 1.25f * 0.25f / (float)b64;   // C2 * (1+COST) / (B*64)
    lgcn_vq_kernel<<<(B_ / 16 + 7) / 8, 256, 0, stream>>>(acc, pcb, cn, cb, items, outi, lossPtr,
                                                          U_, ntiles, B_, lossScale);
}